// DotProductAttention_83476984365340
// MI455X (gfx1250) — compile-verified
//
#include <hip/hip_runtime.h>
#include <hip/hip_bf16.h>

// DotProductAttention — MI455X (gfx1250) optimized.
//
// Key insight: the reference softmaxes over a SINGLETON axis ([B,T,1], axis=2),
// which is identically 1.0. Hence context[b,d] = sum_t values[b,t,d]; query,
// keys and W are mathematically irrelevant. Optimal kernel = streaming f32
// reduction of values (512 MiB @ 23.3 TB/s ~= 22 us), implemented as a
// ones-matrix WMMA accumulation (V_WMMA_F32_16X16X4_F32):
//   D = A(16x4, all ones) x B(4x16 chunk of values) + C
// Every row of D holds the column sums; we read row M=0 at the end.
//
// Layout (per CDNA5 ISA 7.12.2, f32 4x16 B-matrix, wave32):
//   lane 0..15  : column n = lane,    VGPR0 = row t,   VGPR1 = row t+1
//   lane 16..31 : column n = lane-16, VGPR0 = row t+2, VGPR1 = row t+3
// (Since A is all-ones the K<->slot mapping is irrelevant to the sum; only the
//  lane->column fold N = lane & 15 matters, which is forced for a 4x16 B.)
//
// C/D layout: VGPR0, lanes 0..15 = (M=0, N=lane)  -> that's our output row.

typedef __attribute__((ext_vector_type(2))) float v2f;
typedef __attribute__((ext_vector_type(8))) float v8f;

#define B_DIM 32
#define T_DIM 4096
#define D_DIM 1024

__global__ __launch_bounds__(256)
void DotProductAttention_values_sum_wmma(const float* __restrict__ values,
                                         float* __restrict__ out) {
    const int wave = threadIdx.x >> 5;   // 8 waves per block
    const int lane = threadIdx.x & 31;
    const int half = lane >> 4;          // 0: rows t,t+1   1: rows t+2,t+3
    const int col  = lane & 15;          // column within the 16-wide d-tile

    const int b  = blockIdx.y;                       // batch
    const int d0 = blockIdx.x * 128 + wave * 16;     // d-tile base

    // Pointer to values[b][t = 2*half][d0 + col]; second B row is +D.
    const float* p = values
                   + (size_t)b * T_DIM * D_DIM
                   + (size_t)(2 * half) * D_DIM
                   + (size_t)(d0 + col);

    const v2f ones = {1.0f, 1.0f};  // A matrix: all lanes, both VGPRs = 1.0
    v8f c = {};                     // accumulator C/D

    #pragma unroll 4
    for (int t = 0; t < T_DIM; t += 4) {
        v2f bm;
        // Streaming data, no reuse: non-temporal loads keep L2 clean.
        bm.x = __builtin_nontemporal_load(p);
        bm.y = __builtin_nontemporal_load(p + D_DIM);
        // 8-arg form: (neg_a, A, neg_b, B, c_mod, C, reuse_a, reuse_b)
        c = __builtin_amdgcn_wmma_f32_16x16x4_f32(
                false, ones, false, bm, (short)0, c, false, false);
        p += (size_t)4 * D_DIM;
    }

    // Row M=0 of D lives in accumulator element 0 of lanes 0..15.
    if (half == 0) {
        out[(size_t)b * D_DIM + d0 + col] = c[0];
    }
}

extern "C" void kernel_launch(void* const* d_in, const int* in_sizes, int n_in,
                              void* d_out, int out_size, void* d_ws, size_t ws_size,
                              hipStream_t stream) {
    // Input order per setup_inputs(): 0=query [B,D], 1=keys [B,T,D],
    // 2=values [B,T,D], 3=W [T,1]. Only `values` affects the output
    // (softmax over a singleton axis is identically 1).
    const float* values = (const float*)d_in[2];
    float* out = (float*)d_out;     // [B, D] f32

    (void)in_sizes; (void)n_in; (void)d_ws; (void)ws_size; (void)out_size;

    dim3 grid(D_DIM / 128, B_DIM);  // (8, 32) blocks
    DotProductAttention_values_sum_wmma<<<grid, 256, 0, stream>>>(values, out);
}